// SafeDrugLayer_49417893707861
// MI455X (gfx1250) — compile-verified
//
#include <hip/hip_runtime.h>
#include <hip/hip_bf16.h>

#define Bsz  256
#define Ssz  32
#define Isz  128
#define Hsz  256
#define LBL  200
#define NSB  491
#define NAT  8000
#define NMOLc 600

typedef float v2f __attribute__((ext_vector_type(2)));
typedef float v8f __attribute__((ext_vector_type(8)));

// Async global->LDS 16-byte copy (gfx1250 GLOBAL_LOAD_ASYNC_TO_LDS_B128,
// tracked by ASYNCcnt). dst: LDS byte offset, src: 64-bit global address.
static __device__ __forceinline__ void async_b128(const float* g, float* l) {
    unsigned loff = (unsigned)(unsigned long long)
                    (__attribute__((address_space(3))) float*)l;
    asm volatile("global_load_async_to_lds_b128 %0, %1, off"
                 :: "v"(loff), "v"(g) : "memory");
}

// ---------------------------------------------------------------------------
// Async double-buffered fp32 WMMA GEMM (B stored [K,N], aligned dims only):
//   requires M%64==0, N%64==0, K%16==0, 16B-aligned A/B, lda%4==0, ldb%4==0.
// Block = 128 threads = 4 waves (2x2 over a 64x64 C block); each wave owns a
// 32x32 C tile (2x2 WMMA tiles). Per 16-K chunk: 8 async b128 copies stage the
// NEXT chunk's A/B tiles directly into LDS while 16 V_WMMA_F32_16X16X4_F32
// consume the current one; s_wait_asynccnt 8 (in-order) gates buffer reuse.
// ---------------------------------------------------------------------------
__global__ void gemm_wmma_f32_async(const float* __restrict__ A, int lda,
                                    const float* __restrict__ Bm, int ldb,
                                    float* __restrict__ C, int ldc,
                                    int M, int N, int K,
                                    const float* __restrict__ bias,
                                    const float* __restrict__ resid, int ldr,
                                    int act)
{
    __shared__ __align__(16) float As[4][2][32][20];
    __shared__ __align__(16) float Bs[4][2][16][32];

    const int lane = threadIdx.x & 31;
    const int wave = threadIdx.x >> 5;
    const int g    = lane >> 4;
    const int mn   = lane & 15;
    const int tm   = blockIdx.y * 64 + (wave & 1) * 32;
    const int tn   = blockIdx.x * 64 + (wave >> 1) * 32;
    const int kr   = lane & 15, h = lane >> 4;

    const float* arow = A + (long)(tm + lane) * lda;

    auto stage = [&](int buf, int k0) {
#pragma unroll
        for (int q = 0; q < 4; ++q) {
            async_b128(arow + k0 + q * 4, &As[wave][buf][lane][q * 4]);
            async_b128(Bm + (long)(k0 + kr) * ldb + tn + h * 16 + q * 4,
                       &Bs[wave][buf][kr][h * 16 + q * 4]);
        }
    };

    v8f acc[2][2];
#pragma unroll
    for (int mi = 0; mi < 2; ++mi)
#pragma unroll
        for (int ni = 0; ni < 2; ++ni)
            acc[mi][ni] = (v8f){0.f,0.f,0.f,0.f,0.f,0.f,0.f,0.f};

    stage(0, 0);
    int buf = 0;
    for (int k0 = 0; k0 < K; k0 += 16) {
        const bool havenext = (k0 + 16) < K;
        if (havenext) {
            stage(buf ^ 1, k0 + 16);
            asm volatile("s_wait_asynccnt 0x8" ::: "memory");
        } else {
            asm volatile("s_wait_asynccnt 0x0" ::: "memory");
        }
#pragma unroll
        for (int ks = 0; ks < 16; ks += 4) {
            v2f a0, a1, b0, b1;
#pragma unroll
            for (int j = 0; j < 2; ++j) {
                const int kk = ks + j + 2 * g;
                a0[j] = As[wave][buf][mn][kk];
                a1[j] = As[wave][buf][16 + mn][kk];
                b0[j] = Bs[wave][buf][kk][mn];
                b1[j] = Bs[wave][buf][kk][16 + mn];
            }
            acc[0][0] = __builtin_amdgcn_wmma_f32_16x16x4_f32(
                            false, a0, false, b0, (short)0, acc[0][0], false, false);
            acc[0][1] = __builtin_amdgcn_wmma_f32_16x16x4_f32(
                            false, a0, false, b1, (short)0, acc[0][1], false, false);
            acc[1][0] = __builtin_amdgcn_wmma_f32_16x16x4_f32(
                            false, a1, false, b0, (short)0, acc[1][0], false, false);
            acc[1][1] = __builtin_amdgcn_wmma_f32_16x16x4_f32(
                            false, a1, false, b1, (short)0, acc[1][1], false, false);
        }
        buf ^= 1;
    }

#pragma unroll
    for (int mi = 0; mi < 2; ++mi)
#pragma unroll
        for (int ni = 0; ni < 2; ++ni)
#pragma unroll
            for (int j = 0; j < 8; ++j) {
                const int r = tm + mi * 16 + j + 8 * g;
                const int c = tn + ni * 16 + mn;
                float v = acc[mi][ni][j];
                if (bias)  v += bias[c];
                if (resid) v += resid[(long)r * ldr + c];
                if (act == 1)      v = fmaxf(v, 0.f);
                else if (act == 2) v = 1.0f / (1.0f + expf(-v));
                C[(long)r * ldc + c] = v;
            }
}

// ---------------------------------------------------------------------------
// Synchronous generic fp32 WMMA GEMM (TRANSB=1 and/or ragged dims):
//   TRANSB==0 : B stored [K,N];  TRANSB==1 : B stored [N,K]  -> C = A @ B^T
//   EDGE==1   : arbitrary M,N,K via clamped addresses * {0,1} mask
//               (loads always execute -> no EXEC divergence).
// ---------------------------------------------------------------------------
template<int TRANSB, int EDGE>
__global__ void gemm_wmma_f32(const float* __restrict__ A, int lda,
                              const float* __restrict__ Bm, int ldb,
                              float* __restrict__ C, int ldc,
                              int M, int N, int K,
                              const float* __restrict__ bias,
                              const float* __restrict__ resid, int ldr,
                              int act)
{
    __shared__ __align__(16) float As[4][32][20];
    __shared__ __align__(16) float Bs[4][16][32];

    const int lane = threadIdx.x & 31;
    const int wave = threadIdx.x >> 5;
    const int g    = lane >> 4;
    const int mn   = lane & 15;
    const int tm   = blockIdx.y * 64 + (wave & 1) * 32;
    const int tn   = blockIdx.x * 64 + (wave >> 1) * 32;

    v8f acc[2][2];
#pragma unroll
    for (int mi = 0; mi < 2; ++mi)
#pragma unroll
        for (int ni = 0; ni < 2; ++ni)
            acc[mi][ni] = (v8f){0.f,0.f,0.f,0.f,0.f,0.f,0.f,0.f};

    for (int k0 = 0; k0 < K; k0 += 16) {
        // ---------------- stage A tile ----------------
        if (EDGE) {
            const int  row  = tm + lane;
            const int  rowc = row < M ? row : 0;
            const float fr  = row < M ? 1.f : 0.f;
            const float* src = A + (long)rowc * lda;
#pragma unroll
            for (int j = 0; j < 16; ++j) {
                const int kk  = k0 + j;
                const int kkc = kk < K ? kk : 0;
                const float fk = kk < K ? fr : 0.f;
                As[wave][lane][j] = src[kkc] * fk;
            }
        } else {
            const float4* s4 = (const float4*)(A + (long)(tm + lane) * lda + k0);
#pragma unroll
            for (int q = 0; q < 4; ++q)
                *(float4*)&As[wave][lane][q * 4] = s4[q];
        }
        // ---------------- stage B tile -> Bs[k][n] ----------------
        if (TRANSB) {
            if (EDGE) {
                const int  col  = tn + lane;
                const int  colc = col < N ? col : 0;
                const float fc  = col < N ? 1.f : 0.f;
                const float* src = Bm + (long)colc * ldb;
#pragma unroll
                for (int j = 0; j < 16; ++j) {
                    const int kk  = k0 + j;
                    const int kkc = kk < K ? kk : 0;
                    const float fk = kk < K ? fc : 0.f;
                    Bs[wave][j][lane] = src[kkc] * fk;
                }
            } else {
                const float4* s4 = (const float4*)(Bm + (long)(tn + lane) * ldb + k0);
#pragma unroll
                for (int q = 0; q < 4; ++q) {
                    const float4 vq = s4[q];
                    Bs[wave][q * 4 + 0][lane] = vq.x;
                    Bs[wave][q * 4 + 1][lane] = vq.y;
                    Bs[wave][q * 4 + 2][lane] = vq.z;
                    Bs[wave][q * 4 + 3][lane] = vq.w;
                }
            }
        } else {
            const int kr = lane & 15, h = lane >> 4;
            const int  kk  = k0 + kr;
            const int  kkc = kk < K ? kk : 0;
            const float fk = kk < K ? 1.f : 0.f;
            const float* src = Bm + (long)kkc * ldb;
#pragma unroll
            for (int j = 0; j < 16; ++j) {
                const int col  = tn + h * 16 + j;
                const int colc = col < N ? col : 0;
                const float fc = col < N ? fk : 0.f;
                Bs[wave][kr][h * 16 + j] = src[colc] * fc;
            }
        }

        // ---------------- 16 WMMAs ----------------
#pragma unroll
        for (int ks = 0; ks < 16; ks += 4) {
            v2f a0, a1, b0, b1;
#pragma unroll
            for (int j = 0; j < 2; ++j) {
                const int kk = ks + j + 2 * g;
                a0[j] = As[wave][mn][kk];
                a1[j] = As[wave][16 + mn][kk];
                b0[j] = Bs[wave][kk][mn];
                b1[j] = Bs[wave][kk][16 + mn];
            }
            acc[0][0] = __builtin_amdgcn_wmma_f32_16x16x4_f32(
                            false, a0, false, b0, (short)0, acc[0][0], false, false);
            acc[0][1] = __builtin_amdgcn_wmma_f32_16x16x4_f32(
                            false, a0, false, b1, (short)0, acc[0][1], false, false);
            acc[1][0] = __builtin_amdgcn_wmma_f32_16x16x4_f32(
                            false, a1, false, b0, (short)0, acc[1][0], false, false);
            acc[1][1] = __builtin_amdgcn_wmma_f32_16x16x4_f32(
                            false, a1, false, b1, (short)0, acc[1][1], false, false);
        }
    }

#pragma unroll
    for (int mi = 0; mi < 2; ++mi)
#pragma unroll
        for (int ni = 0; ni < 2; ++ni)
#pragma unroll
            for (int j = 0; j < 8; ++j) {
                const int r = tm + mi * 16 + j + 8 * g;
                const int c = tn + ni * 16 + mn;
                if (!EDGE || (r < M && c < N)) {
                    float v = acc[mi][ni][j];
                    if (bias)  v += bias[c];
                    if (resid) v += resid[(long)r * ldr + c];
                    if (act == 1)      v = fmaxf(v, 0.f);
                    else if (act == 2) v = 1.0f / (1.0f + expf(-v));
                    C[(long)r * ldc + c] = v;
                }
            }
}

static inline void gemm(hipStream_t s, const float* A, int lda,
                        const float* B, int ldb, int transB,
                        float* C, int ldc, int M, int N, int K,
                        const float* bias, const float* resid, int ldr, int act)
{
    dim3 grid((N + 63) / 64, (M + 63) / 64);
    const bool edge = (M % 64) || (N % 64) || (K % 16);
    if (transB) {
        if (edge) gemm_wmma_f32<1,1><<<grid,128,0,s>>>(A,lda,B,ldb,C,ldc,M,N,K,bias,resid,ldr,act);
        else      gemm_wmma_f32<1,0><<<grid,128,0,s>>>(A,lda,B,ldb,C,ldc,M,N,K,bias,resid,ldr,act);
    } else {
        if (edge) gemm_wmma_f32<0,1><<<grid,128,0,s>>>(A,lda,B,ldb,C,ldc,M,N,K,bias,resid,ldr,act);
        else      gemm_wmma_f32_async<<<grid,128,0,s>>>(A,lda,B,ldb,C,ldc,M,N,K,bias,resid,ldr,act);
    }
}

// ---------------------------------------------------------------------------
// Small elementwise / utility kernels
// ---------------------------------------------------------------------------
__global__ void zero_k(float* p, int n) {
    int i = blockIdx.x * blockDim.x + threadIdx.x;
    if (i < n) p[i] = 0.f;
}

__global__ void last_idx_k(const int* __restrict__ mask, int* __restrict__ idx) {
    int b = threadIdx.x;
    if (b < Bsz) {
        int s = 0;
        for (int t = 0; t < Ssz; ++t) s += mask[b * Ssz + t];
        idx[b] = s - 1;
    }
}

// torch GRU cell (gates r,z,n), in-place h update; capture h at t==idx[b]
__global__ void gru_gate_k(const float* __restrict__ xp,
                           const float* __restrict__ gm,
                           float* __restrict__ h,
                           const int* __restrict__ idx, int t,
                           float* __restrict__ qin, int qoff)
{
    int i = blockIdx.x * blockDim.x + threadIdx.x;
    if (i >= Bsz * Hsz) return;
    int b = i / Hsz, n = i % Hsz;
    const float* x  = xp + (long)b * 3 * Hsz;
    const float* gg = gm + (long)b * 3 * Hsz;
    float r  = 1.0f / (1.0f + expf(-(x[n]       + gg[n])));
    float z  = 1.0f / (1.0f + expf(-(x[Hsz + n] + gg[Hsz + n])));
    float nn = tanhf(x[2 * Hsz + n] + r * gg[2 * Hsz + n]);
    float hn = (1.0f - z) * nn + z * h[i];
    h[i] = hn;
    if (idx[b] == t) qin[b * (2 * Hsz) + qoff + n] = hn;
}

__global__ void relu_k(float* p, int n) {
    int i = blockIdx.x * blockDim.x + threadIdx.x;
    if (i < n) p[i] = fmaxf(p[i], 0.f);
}

__global__ void gather_k(const float* __restrict__ emb, const int* __restrict__ fp,
                         float* __restrict__ v) {
    int i = blockIdx.x * blockDim.x + threadIdx.x;
    if (i >= NAT * Hsz) return;
    int a = i / Hsz, h = i % Hsz;
    v[i] = emb[(long)fp[a] * Hsz + h];
}

__global__ void seg_bounds_k(const int* __restrict__ seg, int* st, int* en) {
    int i = blockIdx.x * blockDim.x + threadIdx.x;
    if (i >= NAT) return;
    int s = seg[i];
    if (i == 0 || seg[i - 1] != s) st[s] = i;
    if (i == NAT - 1 || seg[i + 1] != s) en[s] = i + 1;
}

// ordered (deterministic) segment sum: one block per molecule, thread = feature
__global__ void seg_sum_k(const float* __restrict__ v, const int* __restrict__ st,
                          const int* __restrict__ en, float* __restrict__ mol) {
    int m = blockIdx.x, h = threadIdx.x;
    float acc = 0.f;
    int a0 = st[m], a1 = en[m];
    for (int a = a0; a < a1; ++a) acc += v[(long)a * Hsz + h];
    mol[(long)m * Hsz + h] = acc;
}

__global__ void maskmul_k(const float* __restrict__ wm, const float* __restrict__ ddi,
                          float* __restrict__ o) {
    int i = blockIdx.x * blockDim.x + threadIdx.x;
    if (i >= NSB * LBL) return;
    int s = i / LBL, l = i % LBL;
    o[i] = wm[i] * ddi[(long)l * NSB + s];   // w_masklin * ddi_mask_H^T
}

__global__ void layernorm_k(const float* __restrict__ x, const float* __restrict__ gamma,
                            const float* __restrict__ beta, float* __restrict__ y) {
    __shared__ float red[256];
    int b = blockIdx.x, t = threadIdx.x;
    float v = (t < LBL) ? x[(long)b * LBL + t] : 0.f;
    red[t] = v; __syncthreads();
    for (int s = 128; s > 0; s >>= 1) { if (t < s) red[t] += red[t + s]; __syncthreads(); }
    float mu = red[0] * (1.0f / LBL);
    __syncthreads();
    float d = (t < LBL) ? (v - mu) : 0.f;
    red[t] = d * d; __syncthreads();
    for (int s = 128; s > 0; s >>= 1) { if (t < s) red[t] += red[t + s]; __syncthreads(); }
    float var = red[0] * (1.0f / LBL);
    if (t < LBL)
        y[(long)b * LBL + t] = d * rsqrtf(var + 1e-5f) * gamma[t] + beta[t];
}

__global__ void mul_k(const float* __restrict__ a, const float* __restrict__ b,
                      float* __restrict__ o, int n) {
    int i = blockIdx.x * blockDim.x + threadIdx.x;
    if (i < n) o[i] = a[i] * b[i];
}

// ---------------------------------------------------------------------------
extern "C" void kernel_launch(void* const* d_in, const int* in_sizes, int n_in,
                              void* d_out, int out_size, void* d_ws, size_t ws_size,
                              hipStream_t stream)
{
    const float* xc       = (const float*)d_in[0];
    const float* xpr      = (const float*)d_in[1];
    const int*   mask     = (const int*)  d_in[2];
    const float* w_ih_c   = (const float*)d_in[3];
    const float* w_hh_c   = (const float*)d_in[4];
    const float* b_ih_c   = (const float*)d_in[5];
    const float* b_hh_c   = (const float*)d_in[6];
    const float* w_ih_p   = (const float*)d_in[7];
    const float* w_hh_p   = (const float*)d_in[8];
    const float* b_ih_p   = (const float*)d_in[9];
    const float* b_hh_p   = (const float*)d_in[10];
    const float* w_query  = (const float*)d_in[11];
    const float* b_query  = (const float*)d_in[12];
    const float* w_bip    = (const float*)d_in[13];
    const float* b_bip    = (const float*)d_in[14];
    const float* w_mlin   = (const float*)d_in[15];
    const float* ddi      = (const float*)d_in[16];
    const float* embed_fp = (const float*)d_in[17];
    const int*   fing     = (const int*)  d_in[18];
    const float* adj      = (const float*)d_in[19];
    const int*   seg      = (const int*)  d_in[20];
    const float* w_g0     = (const float*)d_in[21];
    const float* b_g0     = (const float*)d_in[22];
    const float* w_g1     = (const float*)d_in[23];
    const float* b_g1     = (const float*)d_in[24];
    const float* avg      = (const float*)d_in[25];
    const float* w_out    = (const float*)d_in[26];
    const float* b_out    = (const float*)d_in[27];
    const float* ln_g     = (const float*)d_in[28];
    const float* ln_b     = (const float*)d_in[29];
    float* out = (float*)d_out;

    // ---- carve workspace ----
    char* pw = (char*)d_ws;
    auto alloc = [&](size_t nfloats) -> float* {
        float* r = (float*)pw;
        pw += ((nfloats * 4 + 255) / 256) * 256;
        return r;
    };
    float* hbuf   = alloc(2 * Bsz * Hsz);       // h_c | h_p
    float* h_c = hbuf, *h_p = hbuf + Bsz * Hsz;
    float* xpc    = alloc(Bsz * 3 * Hsz);
    float* xpp    = alloc(Bsz * 3 * Hsz);
    float* gc     = alloc(Bsz * 3 * Hsz);
    float* gp     = alloc(Bsz * 3 * Hsz);
    float* qin    = alloc(Bsz * 2 * Hsz);
    float* query  = alloc(Bsz * Hsz);
    float* v0     = alloc((size_t)NAT * Hsz);
    float* v1     = alloc((size_t)NAT * Hsz);
    float* v2     = alloc((size_t)NAT * Hsz);
    float* mol    = alloc((size_t)NMOLc * Hsz);
    float* emb    = alloc((size_t)LBL * Hsz);
    float* match  = alloc((size_t)Bsz * LBL);
    float* x2     = alloc((size_t)Bsz * LBL);
    float* att    = alloc((size_t)Bsz * LBL);
    float* bip    = alloc((size_t)Bsz * NSB);
    float* Wm     = alloc((size_t)NSB * LBL);
    float* bipatt = alloc((size_t)Bsz * LBL);
    int*   idx    = (int*)alloc(256);
    int*   st     = (int*)alloc(NMOLc);
    int*   en     = (int*)alloc(NMOLc);
    (void)ws_size; (void)in_sizes; (void)n_in; (void)out_size;

    // ---- init ----
    zero_k<<<(2 * Bsz * Hsz + 255) / 256, 256, 0, stream>>>(hbuf, 2 * Bsz * Hsz);
    last_idx_k<<<1, 256, 0, stream>>>(mask, idx);

    // ---- two GRUs, 32 sequential steps ----
    for (int t = 0; t < Ssz; ++t) {
        gemm(stream, xc  + (long)t * Isz, Ssz * Isz, w_ih_c, Isz, 1,
             xpc, 3 * Hsz, Bsz, 3 * Hsz, Isz, b_ih_c, nullptr, 0, 0);
        gemm(stream, xpr + (long)t * Isz, Ssz * Isz, w_ih_p, Isz, 1,
             xpp, 3 * Hsz, Bsz, 3 * Hsz, Isz, b_ih_p, nullptr, 0, 0);
        gemm(stream, h_c, Hsz, w_hh_c, Hsz, 1, gc, 3 * Hsz,
             Bsz, 3 * Hsz, Hsz, b_hh_c, nullptr, 0, 0);
        gemm(stream, h_p, Hsz, w_hh_p, Hsz, 1, gp, 3 * Hsz,
             Bsz, 3 * Hsz, Hsz, b_hh_p, nullptr, 0, 0);
        gru_gate_k<<<(Bsz * Hsz + 255) / 256, 256, 0, stream>>>(xpc, gc, h_c, idx, t, qin, 0);
        gru_gate_k<<<(Bsz * Hsz + 255) / 256, 256, 0, stream>>>(xpp, gp, h_p, idx, t, qin, Hsz);
    }

    // queries = relu(patient_last) @ w_query + b_query
    relu_k<<<(Bsz * 2 * Hsz + 255) / 256, 256, 0, stream>>>(qin, Bsz * 2 * Hsz);
    gemm(stream, qin, 2 * Hsz, w_query, Hsz, 0, query, Hsz,
         Bsz, Hsz, 2 * Hsz, b_query, nullptr, 0, 0);

    // ---- MPNN over molecular graph ----
    gather_k<<<(NAT * Hsz + 255) / 256, 256, 0, stream>>>(embed_fp, fing, v0);
    gemm(stream, v0, Hsz, w_g0, Hsz, 0, v1, Hsz, NAT, Hsz, Hsz, b_g0, nullptr, 0, 1);
    gemm(stream, adj, NAT, v1, Hsz, 0, v2, Hsz, NAT, Hsz, NAT, nullptr, v1, Hsz, 0);
    gemm(stream, v2, Hsz, w_g1, Hsz, 0, v1, Hsz, NAT, Hsz, Hsz, b_g1, nullptr, 0, 1);
    gemm(stream, adj, NAT, v1, Hsz, 0, v0, Hsz, NAT, Hsz, NAT, nullptr, v1, Hsz, 0);

    seg_bounds_k<<<(NAT + 255) / 256, 256, 0, stream>>>(seg, st, en);
    seg_sum_k<<<NMOLc, Hsz, 0, stream>>>(v0, st, en, mol);

    gemm(stream, avg, NMOLc, mol, Hsz, 0, emb, Hsz, LBL, Hsz, NMOLc, nullptr, nullptr, 0, 0);
    gemm(stream, query, Hsz, emb, Hsz, 1, match, LBL, Bsz, LBL, Hsz, nullptr, nullptr, 0, 2);
    gemm(stream, match, LBL, w_out, LBL, 0, x2, LBL, Bsz, LBL, LBL, b_out, match, LBL, 0);
    layernorm_k<<<Bsz, 256, 0, stream>>>(x2, ln_g, ln_b, att);

    gemm(stream, query, Hsz, w_bip, NSB, 0, bip, NSB, Bsz, NSB, Hsz, b_bip, nullptr, 0, 0);
    maskmul_k<<<(NSB * LBL + 255) / 256, 256, 0, stream>>>(w_mlin, ddi, Wm);
    gemm(stream, bip, NSB, Wm, LBL, 0, bipatt, LBL, Bsz, LBL, NSB, nullptr, nullptr, 0, 0);

    mul_k<<<(Bsz * LBL + 255) / 256, 256, 0, stream>>>(bipatt, att, out, Bsz * LBL);
}